// RBFController_65395172049491
// MI455X (gfx1250) — compile-verified
//
#include <hip/hip_runtime.h>
#include <hip/hip_bf16.h>
#include <math.h>

typedef __attribute__((ext_vector_type(2))) float v2f;
typedef __attribute__((ext_vector_type(8))) float v8f;
typedef __attribute__((ext_vector_type(4))) unsigned int v4u;
typedef __attribute__((ext_vector_type(4))) int v4i;
typedef __attribute__((ext_vector_type(8))) int v8i;

#define B_BATCH 2048
#define N_C 32
#define D_DIM 64
#define A_DIM 8

// ---- shared memory layout (floats) -------------------------------------
#define BM_OFF   0            // 64x65 = 4160  (B_mat -> Cholesky L1; reused: G + Cross)
#define BQ_OFF   4160         // 64x65 = 4160  (TDM lands raw cov here -> B_q -> L2; reused: Q, R, AC, P)
#define V_OFF    8320         // 64x33 = 2112  (S^T -> V = L1^{-1} S^T -> T2 = L1^{-T} V)
#define W_OFF    10432        // 64x34 = 2176  (cols 0..31: C^T -> W = L2^{-1} C^T ; col 32: mean -> w)
#define MISC_OFF 12608
#define SMEM_FLOATS (12608 + 552)

#define BM(i,j)  smem[BM_OFF + (i)*65 + (j)]
#define BQ(i,j)  smem[BQ_OFF + (i)*65 + (j)]
#define VX(d,n)  smem[V_OFF  + (d)*33 + (n)]
#define WX(d,n)  smem[W_OFF  + (d)*34 + (n)]
// overlays (live after the base regions are dead):
#define G_(n,m)  smem[BM_OFF + (n)*33 + (m)]            // 32x33 = 1056
#define CR(d,c)  smem[BM_OFF + 1056 + (d)*17 + (c)]     // 64x17 = 1088
#define QX(n,m)  smem[BQ_OFF + (n)*33 + (m)]            // 32x33 = 1056
#define RX(n,a)  smem[BQ_OFF + 1056 + (n)*9 + (a)]      // 32x9  = 288
#define ACX(a,c) smem[BQ_OFF + 1344 + (a)*8 + (c)]      // 64
#define PX(n,a)  smem[BQ_OFF + 1408 + (n)*16 + (a)]     // 32x16 = 512

// exp1[n,m] = -0.25 * sum_d (c_n[d]-c_m[d])^2 / l[d]^2 ; ws[1024] = sum_d log(l^2/2)
__global__ void rbf_precompute(const float* __restrict__ centers,
                               const float* __restrict__ ls,
                               float* __restrict__ ws) {
    __shared__ float sC[N_C][D_DIM];
    __shared__ float sIl2[D_DIM];
    int tid = threadIdx.x;
    for (int i = tid; i < N_C * D_DIM; i += 256) sC[i >> 6][i & 63] = centers[i];
    if (tid < D_DIM) { float l = ls[tid]; sIl2[tid] = 1.0f / (l * l); }
    __syncthreads();
    for (int p = tid; p < N_C * N_C; p += 256) {
        int n = p >> 5, m = p & 31;
        float s = 0.0f;
        for (int d = 0; d < D_DIM; ++d) {
            float df = sC[n][d] - sC[m][d];
            s += df * df * sIl2[d];
        }
        ws[p] = -0.25f * s;
    }
    if (tid == 0) {
        float s = 0.0f;
        for (int d = 0; d < D_DIM; ++d) { float l = ls[d]; s += logf(0.5f * l * l); }
        ws[1024] = s;
    }
}

__global__ __launch_bounds__(256)
void rbf_main(const float* __restrict__ mean,
              const float* __restrict__ cov,
              const float* __restrict__ centers,
              const float* __restrict__ weights,
              const float* __restrict__ ls,
              const float* __restrict__ ws,
              float* __restrict__ out_mean,
              float* __restrict__ out_cov,
              float* __restrict__ out_cross) {
    __shared__ float smem[SMEM_FLOATS];
    float* sMean = &smem[MISC_OFF + 0];     // 64
    float* sInvL = &smem[MISC_OFF + 64];    // 64
    float* sL2h  = &smem[MISC_OFF + 128];   // 64
    float* sPhi  = &smem[MISC_OFF + 192];   // 32
    float* sU    = &smem[MISC_OFF + 224];   // 32
    float* sAm   = &smem[MISC_OFF + 256];   // 8  (raw action mean)
    float* sDc   = &smem[MISC_OFF + 264];   // 8
    float* sSm   = &smem[MISC_OFF + 272];   // 8  (squashed mean)
    float* sCv   = &smem[MISC_OFF + 280];   // 8  (squash cross diag)
    float* sScal = &smem[MISC_OFF + 288];   // 8  [0]=logdetBm [1]=logdetBq [2]=s [3]=c_q [4],[5]=chol tmp
    float* sWt   = &smem[MISC_OFF + 296];   // 32x8

    const int b = blockIdx.x;
    const int tid = threadIdx.x;
    const float* covb = cov + (size_t)b * (D_DIM * D_DIM);

    // ---- TDM: DMA cov[b] (64x64 f32) directly into LDS (BQ region),
    //      padding 1 dword after every 64 to match the stride-65 layout.
    if (tid < 32) {  // one wave issues the tensor DMA; EXEC is ignored by TDM
        unsigned long long ga = (unsigned long long)(uintptr_t)covb;
        unsigned lds_addr = (unsigned)(uintptr_t)(&smem[BQ_OFF]);
        v4u g0;
        g0[0] = 1u;                                   // count=1, user descriptor
        g0[1] = lds_addr;                             // lds_addr[31:0]
        g0[2] = (unsigned)(ga & 0xFFFFFFFFu);         // global_addr[31:0]
        g0[3] = (unsigned)((ga >> 32) & 0x01FFFFFFu)  // global_addr[56:32]
                | 0x80000000u;                        // type=2 ("image") at [127:126]
        v8i g1;
        g1[0] = (int)((2u << 16)        // data_size = 4 bytes
                    | (1u << 20)        // pad_enable
                    | (5u << 22));      // pad_interval: 64 dwords (pad_amount=0 -> 1 dword)
        g1[1] = (int)(64u << 16);       // tensor_dim0 = 64 (low16 at bits 63:48)
        g1[2] = (int)(64u << 16);       // tensor_dim1 = 64 (low16 at bits 95:80)
        g1[3] = (int)(64u << 16);       // tile_dim0  = 64 (bits 127:112)
        g1[4] = 64;                     // tile_dim1  = 64 (bits 143:128), tile_dim2 = 0
        g1[5] = 64;                     // tensor_dim0_stride = 64 (low32)
        g1[6] = 0;
        g1[7] = 0;
        v4i g2 = {0, 0, 0, 0};          // unused (2D tensor)
        v4i g3 = {0, 0, 0, 0};
#if defined(__clang_major__) && __clang_major__ >= 23
        v8i g4 = {0, 0, 0, 0, 0, 0, 0, 0};
        __builtin_amdgcn_tensor_load_to_lds(g0, g1, g2, g3, g4, 0);
#else
        __builtin_amdgcn_tensor_load_to_lds(g0, g1, g2, g3, 0);
#endif
    }

    __builtin_prefetch(centers, 0, 1);  // global_prefetch_b8 (small, L2-resident)

    if (tid < 64) {
        float l = ls[tid];
        sInvL[tid] = 1.0f / l;
        sL2h[tid]  = 0.5f * l * l;
        sMean[tid] = mean[(size_t)b * 64 + tid];
    }
    sWt[tid] = weights[tid];  // 256 threads == 32*8 weights
    __syncthreads();

    // Stage the two RHS blocks into LDS (overlaps with the tensor DMA)
    for (int idx = tid; idx < 2048; idx += 256) {
        int n = idx >> 6, d = idx & 63;
        float c = centers[idx];
        VX(d, n) = (c - sMean[d]) * sInvL[d];   // S^T (scaled inputs)
        WX(d, n) = c;                           // C^T
    }
    if (tid < 64) WX(tid, 32) = sMean[tid];

    if (tid < 32) __builtin_amdgcn_s_wait_tensorcnt(0);  // cov tile landed in LDS
    __syncthreads();

    // Build B_mat from the LDS cov tile; bump B_q's diagonal in place
    for (int idx = tid; idx < 4096; idx += 256) {
        int i = idx >> 6, j = idx & 63;
        float cv = BQ(i, j);                    // raw cov from TDM
        BM(i, j) = cv * sInvL[i] * sInvL[j] + ((i == j) ? 1.0f : 0.0f);
        if (i == j) BQ(i, i) = cv + sL2h[i];
    }
    __syncthreads();

    // ---- Cholesky of both matrices (lower triangular, in place) --------
    for (int k = 0; k < 64; ++k) {
        if (tid == 0) { float v = BM(k, k); float s = sqrtf(v); BM(k, k) = s; sScal[4] = 1.0f / s; }
        if (tid == 1) { float v = BQ(k, k); float s = sqrtf(v); BQ(k, k) = s; sScal[5] = 1.0f / s; }
        __syncthreads();
        if (tid < 128) {
            int m = tid >> 6, i = tid & 63;
            if (i > k) {
                if (m == 0) BM(i, k) *= sScal[4];
                else        BQ(i, k) *= sScal[5];
            }
        }
        __syncthreads();
        // rank-1 update of the trailing LOWER triangle, row-owner mapping (no div/mod)
        if (tid < 128) {
            int m = tid >> 6, i = tid & 63;
            if (i > k) {
                if (m == 0) {
                    float lik = BM(i, k);
                    for (int j = k + 1; j <= i; ++j) BM(i, j) -= lik * BM(j, k);
                } else {
                    float lik = BQ(i, k);
                    for (int j = k + 1; j <= i; ++j) BQ(i, j) -= lik * BQ(j, k);
                }
            }
        }
        __syncthreads();
    }

    // ---- Forward substitution: V = L1^{-1} S^T ; W = L2^{-1} [C^T|mean]
    if (tid < 65) {
        const int isW = (tid >= 32);
        const int c = isW ? (tid - 32) : tid;
        const int l_off = isW ? BQ_OFF : BM_OFF;
        const int x_off = isW ? W_OFF : V_OFF;
        const int xs    = isW ? 34 : 33;
        for (int i = 0; i < 64; ++i) {
            float s = smem[x_off + i * xs + c];
            for (int j = 0; j < i; ++j)
                s -= smem[l_off + i * 65 + j] * smem[x_off + j * xs + c];
            smem[x_off + i * xs + c] = s / smem[l_off + i * 65 + i];
        }
    }
    if (tid == 65) { float s = 0.0f; for (int k = 0; k < 64; ++k) s += logf(BM(k, k)); sScal[0] = 2.0f * s; }
    if (tid == 66) { float s = 0.0f; for (int k = 0; k < 64; ++k) s += logf(BQ(k, k)); sScal[1] = 2.0f * s; }
    __syncthreads();

    // ---- phi, u, s, c_q ------------------------------------------------
    if (tid < 32) {
        float s = 0.0f;
        for (int d = 0; d < 64; ++d) { float v = VX(d, tid); s += v * v; }
        sPhi[tid] = expf(-0.5f * (sScal[0] + s));   // VARIANCE=1
    } else if (tid < 64) {
        int n = tid - 32;
        float s = 0.0f;
        for (int d = 0; d < 64; ++d) s += WX(d, n) * WX(d, 32);
        sU[n] = s;
    } else if (tid == 64) {
        float s = 0.0f;
        for (int d = 0; d < 64; ++d) s += WX(d, 32) * WX(d, 32);
        sScal[2] = s;
    } else if (tid == 65) {
        sScal[3] = expf(0.5f * (ws[1024] - sScal[1]));  // c_q
    }
    __syncthreads();

    // ---- Back substitution: T2 = L1^{-T} V (in place in sV) ; build P --
    if (tid < 32) {
        int c = tid;
        for (int i = 63; i >= 0; --i) {
            float s = VX(i, c);
            for (int j = i + 1; j < 64; ++j)
                s -= BM(j, i) * VX(j, c);
            VX(i, c) = s / BM(i, i);
        }
    }
    if (tid >= 128) {
        for (int p = tid - 128; p < 512; p += 128) {
            int n = p >> 4, a = p & 15;
            PX(n, a) = (a < 8) ? sPhi[n] * sWt[n * 8 + a] : 0.0f;
        }
    }
    if (tid >= 64 && tid < 72) {
        int a = tid - 64;
        float s = 0.0f;
        for (int n = 0; n < 32; ++n) s += sPhi[n] * sWt[n * 8 + a];
        sAm[a] = s;  // raw action mean
    }
    __syncthreads();

    // ---- WMMA stage: waves 0-3: G = W^T W ; waves 4-7: Cross = T2 * P --
    {
        const int wave = tid >> 5;
        const int lane = tid & 31;
        const int half = lane >> 4;
        const int l15  = lane & 15;
        v8f acc = {};
        if (wave < 4) {
            const int m0 = (wave >> 1) * 16;
            const int n0 = (wave & 1) * 16;
            for (int k0 = 0; k0 < 64; k0 += 4) {
                int kb = k0 + 2 * half;
                v2f a, bb;
                a.x  = WX(kb,     m0 + l15);
                a.y  = WX(kb + 1, m0 + l15);
                bb.x = WX(kb,     n0 + l15);
                bb.y = WX(kb + 1, n0 + l15);
                acc = __builtin_amdgcn_wmma_f32_16x16x4_f32(
                        false, a, false, bb, (short)0, acc, false, false);
            }
            for (int r = 0; r < 8; ++r)
                G_(m0 + r + 8 * half, n0 + l15) = acc[r];
        } else {
            const int r0 = (wave - 4) * 16;
            for (int k0 = 0; k0 < 32; k0 += 4) {
                int kb = k0 + 2 * half;
                v2f a, bb;
                a.x  = VX(r0 + l15, kb);
                a.y  = VX(r0 + l15, kb + 1);
                bb.x = PX(kb,     l15);
                bb.y = PX(kb + 1, l15);
                acc = __builtin_amdgcn_wmma_f32_16x16x4_f32(
                        false, a, false, bb, (short)0, acc, false, false);
            }
            for (int r = 0; r < 8; ++r)
                CR(r0 + r + 8 * half, l15) = acc[r];
        }
    }
    __syncthreads();

    // ---- Q matrix ------------------------------------------------------
    for (int p = tid; p < 1024; p += 256) {
        int n = p >> 5, m = p & 31;
        float e2 = -0.5f * (0.25f * (G_(n, n) + G_(m, m)) + 0.5f * G_(n, m)
                            - sU[n] - sU[m] + sScal[2]);
        QX(n, m) = sScal[3] * expf(ws[p] + e2);
    }
    __syncthreads();

    // ---- R = Q * weights ----------------------------------------------
    {
        int n = tid >> 3, a = tid & 7;
        float s = 0.0f;
        for (int m = 0; m < 32; ++m) s += QX(n, m) * sWt[m * 8 + a];
        RX(n, a) = s;
    }
    __syncthreads();

    // ---- action_cov = W^T R - am am^T ; symmetrize ; +1e-6 I -----------
    if (tid < 64) {
        int a = tid >> 3, c = tid & 7;
        float s = 0.0f;
        for (int n = 0; n < 32; ++n) s += sWt[n * 8 + a] * RX(n, c);
        ACX(a, c) = s - sAm[a] * sAm[c];
    }
    __syncthreads();
    float acv = 0.0f;
    if (tid < 64) {
        int a = tid >> 3, c = tid & 7;
        acv = 0.5f * (ACX(a, c) + ACX(c, a));
        if (a == c) acv += 1e-6f;
    }
    __syncthreads();
    if (tid < 64) ACX(tid >> 3, tid & 7) = acv;
    __syncthreads();

    // ---- squash_sin ----------------------------------------------------
    if (tid < 8) {
        int a = tid;
        float dc = ACX(a, a);
        float e = expf(-0.5f * dc);
        sDc[a] = dc;
        sSm[a] = e * sinf(sAm[a]);
        sCv[a] = e * cosf(sAm[a]);
    }
    __syncthreads();

    if (tid < 8) out_mean[(size_t)b * 8 + tid] = sSm[tid];
    if (tid < 64) {
        int a = tid >> 3, c = tid & 7;
        float lq = -0.5f * (sDc[a] + sDc[c]);
        float q = expf(lq);
        float cv = ACX(a, c);
        float sq = 0.5f * ((expf(lq + cv) - q) * cosf(sAm[a] - sAm[c])
                         - (expf(lq - cv) - q) * cosf(sAm[a] + sAm[c]));
        out_cov[(size_t)b * 64 + tid] = sq;
    }
    for (int p = tid; p < 512; p += 256) {
        int d = p >> 3, c = p & 7;
        out_cross[(size_t)b * 512 + p] = CR(d, c) * sInvL[d] * sCv[c];
    }
}

extern "C" void kernel_launch(void* const* d_in, const int* in_sizes, int n_in,
                              void* d_out, int out_size, void* d_ws, size_t ws_size,
                              hipStream_t stream) {
    const float* mean    = (const float*)d_in[0];
    const float* cov     = (const float*)d_in[1];
    const float* centers = (const float*)d_in[2];
    const float* weights = (const float*)d_in[3];
    const float* ls      = (const float*)d_in[4];
    float* out = (float*)d_out;
    float* ws  = (float*)d_ws;

    rbf_precompute<<<1, 256, 0, stream>>>(centers, ls, ws);
    rbf_main<<<B_BATCH, 256, 0, stream>>>(
        mean, cov, centers, weights, ls, ws,
        out,                                          // action_mean  (2048 x 8)
        out + (size_t)B_BATCH * A_DIM,                // action_cov   (2048 x 8 x 8)
        out + (size_t)B_BATCH * A_DIM * (1 + A_DIM)); // cross_cov    (2048 x 64 x 8)
}